// CudaModel_12369505812694
// MI455X (gfx1250) — compile-verified
//
#include <hip/hip_runtime.h>

// Gated DeltaNet single step, output-only formulation:
//   out = g*(q^T S) + beta*(q.k)*(v - g*(k^T S))
// Memory-bound: 805 MB state streamed once -> ~35 us floor at 23.3 TB/s.
// Reductions done with V_WMMA_F32_16X16X4_F32 (f32 precision preserved).

typedef __attribute__((ext_vector_type(2))) float v2f;
typedef __attribute__((ext_vector_type(8))) float v8f;

#define DKC 128
#define DVC 128

__global__ __launch_bounds__(256)
void gdn_step_kernel(const float* __restrict__ q,
                     const float* __restrict__ k,
                     const float* __restrict__ v,
                     const float* __restrict__ beta,
                     const float* __restrict__ gate,
                     const float* __restrict__ state,
                     float* __restrict__ out)
{
    // LDS: [0..127] = zeros (A rows 2..15), [128..255] = q, [256..383] = k
    __shared__ float lds[3 * DKC];

    const int bh   = blockIdx.x;      // one workgroup per (b,h)
    const int tid  = threadIdx.x;
    const int lane = tid & 31;
    const int wave = tid >> 5;        // 8 waves -> 8 dv tiles of 16

    const float* qv = q + (size_t)bh * DKC;
    const float* kv = k + (size_t)bh * DKC;

    if (tid < DKC) { lds[tid] = 0.0f; lds[DKC + tid] = qv[tid]; }
    else           { lds[2 * DKC + (tid - DKC)] = kv[tid - DKC]; }
    __syncthreads();

    const int m  = lane & 15;         // row (A) / column (B) index within tile
    const int kk = (lane >> 4) * 2;   // K sub-offset: lanes 0-15 -> K0/K1, 16-31 -> K2/K3

    // A operand row source: M=0 -> q, M=1 -> k, M>=2 -> zero row (branch-free)
    const int arow = (m == 0) ? DKC : ((m == 1) ? 2 * DKC : 0);

    // B operand: lane reads S[k0 + kk + {0,1}, dv0 + m]; 64B contiguous per half-wave
    const int dv0 = wave * 16;
    const float* sp = state + (size_t)bh * DKC * DVC + (size_t)kk * DVC + (dv0 + m);

    v8f c = {};
    #pragma unroll 4
    for (int k0 = 0; k0 < DKC; k0 += 4) {
        v2f a;
        a.x = lds[arow + k0 + kk];
        a.y = lds[arow + k0 + kk + 1];
        v2f bm;
        bm.x = sp[(size_t)k0 * DVC];
        bm.y = sp[(size_t)k0 * DVC + DVC];
        // D = A(16x4) * B(4x16) + C ; row0 = partial q^T S, row1 = partial k^T S
        c = __builtin_amdgcn_wmma_f32_16x16x4_f32(false, a, false, bm,
                                                  (short)0, c, false, false);
    }

    // qk = dot(q,k): 4 elems per lane + 5-round xor-shuffle reduction (wave32)
    float4 q4 = *(const float4*)&lds[DKC + lane * 4];
    float4 k4 = *(const float4*)&lds[2 * DKC + lane * 4];
    float qk = q4.x * k4.x + q4.y * k4.y + q4.z * k4.z + q4.w * k4.w;
    #pragma unroll
    for (int off = 16; off > 0; off >>= 1)
        qk += __shfl_xor(qk, off, 32);

    const float g  = gate[bh];
    const float bt = beta[bh];

    // D layout: VGPR i, lanes 0-15 = row M=i. Row 0 = qS, row 1 = kS.
    if (lane < 16) {
        const int dv = dv0 + lane;
        const float qs = c[0];
        const float ks = c[1];
        const float vv = v[(size_t)bh * DVC + dv];
        out[(size_t)bh * DVC + dv] = g * qs + bt * qk * (vv - g * ks);
    }
}

extern "C" void kernel_launch(void* const* d_in, const int* in_sizes, int n_in,
                              void* d_out, int out_size, void* d_ws, size_t ws_size,
                              hipStream_t stream) {
    const float* q     = (const float*)d_in[0];
    const float* k     = (const float*)d_in[1];
    const float* v     = (const float*)d_in[2];
    const float* beta  = (const float*)d_in[3];
    const float* gate  = (const float*)d_in[4];
    const float* state = (const float*)d_in[5];
    float* out = (float*)d_out;

    const int BH = in_sizes[3];   // beta is [B,H] -> number of (b,h) pairs
    gdn_step_kernel<<<BH, 256, 0, stream>>>(q, k, v, beta, gate, state, out);
}